// ProteinFeatureExtractor_83416854823220
// MI455X (gfx1250) — compile-verified
//
#include <hip/hip_runtime.h>

#define N_NODES 50000
#define N_EDGES 800000
#define N_GRAPHS 256
#define NODE_DIM 6
#define HIDDEN 128
#define OUT_DIM 512
#define NUM_LAYERS 3

typedef float v2f __attribute__((ext_vector_type(2)));
typedef float v8f __attribute__((ext_vector_type(8)));

// ---- CDNA5 async copy: global -> LDS, 16 bytes per lane, tracked by ASYNCcnt ----
__device__ __forceinline__ void async_ld16(unsigned lds_byte_addr, const float* gaddr) {
    asm volatile("global_load_async_to_lds_b128 %0, %1, off"
                 :: "v"(lds_byte_addr), "v"(gaddr)
                 : "memory");
}
__device__ __forceinline__ void wait_async0() {
    asm volatile("s_wait_asynccnt 0x0" ::: "memory");
}
__device__ __forceinline__ unsigned lds_addr(const void* p) {
    // low 32 bits of a generic pointer into LDS are the LDS byte address
    return (unsigned)(unsigned long long)p;
}

// ---------------- utility ----------------
__global__ void zero_f32(float* __restrict__ p, int n) {
    int i = blockIdx.x * blockDim.x + threadIdx.x;
    if (i < n) p[i] = 0.0f;
}

// ---------------- input projection: h = relu(x @ Wp + bp) ----------------
__global__ void node_proj(const float* __restrict__ x, const float* __restrict__ W,
                          const float* __restrict__ b, float* __restrict__ h) {
    int n = blockIdx.x;
    int c = threadIdx.x;
    float acc = b[c];
#pragma unroll
    for (int k = 0; k < NODE_DIM; ++k)
        acc += x[n * NODE_DIM + k] * W[k * HIDDEN + c];
    h[n * HIDDEN + c] = fmaxf(acc, 0.0f);
}

// ---------------- edge scatter: agg[dst] += h[src] ----------------
// one wave32 per edge; float4 gather + 4x global_atomic_add_f32 per lane
__global__ void edge_scatter(const int* __restrict__ src, const int* __restrict__ dst,
                             const float* __restrict__ h, float* __restrict__ agg) {
    int tid  = blockIdx.x * blockDim.x + threadIdx.x;
    int e    = tid >> 5;
    int lane = tid & 31;
    if (e >= N_EDGES) return;
    int s = src[e];
    int d = dst[e];
    float4 v = *(const float4*)(h + (size_t)s * HIDDEN + lane * 4);
    float* o = agg + (size_t)d * HIDDEN + lane * 4;
    atomicAdd(o + 0, v.x);
    atomicAdd(o + 1, v.y);
    atomicAdd(o + 2, v.z);
    atomicAdd(o + 3, v.w);
}

// ---------------- GIN GEMM: hout = relu((h+agg) @ W + bias) ----------------
// M tile = 16 rows/block, N tile = 16 cols/wave (8 waves), K = 128 in chunks of 32.
// Double-buffered async global->LDS staging overlapped with v_wmma_f32_16x16x4_f32.
__global__ __launch_bounds__(256) void gin_gemm(const float* __restrict__ h,
                                                const float* __restrict__ agg,
                                                const float* __restrict__ W,
                                                const float* __restrict__ bias,
                                                float* __restrict__ hout) {
    // padded strides: 36 floats = 144B, 132 floats = 528B (both multiples of 16B)
    __shared__ __align__(16) float Hs[2][16][36];
    __shared__ __align__(16) float Ag[2][16][36];
    __shared__ __align__(16) float Ws[2][32][132];

    const int tid   = threadIdx.x;
    const int lane  = tid & 31;
    const int wave  = tid >> 5;          // 0..7 -> N tile
    const int row0  = blockIdx.x * 16;
    const int nbase = wave * 16;
    const int m     = lane & 15;
    const int half  = lane >> 4;         // lane group 0/1
    const int kb    = half * 2;          // K sub-offset per ISA A/B striping

    // issue async copies for one 32-wide K chunk into buffer `buf`
    auto issue = [&](int buf, int kc) {
        if (tid < 128) {                 // Hs: 16x32 floats = 128 float4
            int r = tid >> 3, q = (tid & 7) * 4;
            async_ld16(lds_addr(&Hs[buf][r][q]),
                       h + (size_t)(row0 + r) * HIDDEN + kc + q);
        } else {                          // Ag: 16x32 floats = 128 float4
            int t = tid - 128;
            int r = t >> 3, q = (t & 7) * 4;
            async_ld16(lds_addr(&Ag[buf][r][q]),
                       agg + (size_t)(row0 + r) * HIDDEN + kc + q);
        }
#pragma unroll
        for (int j = 0; j < 4; ++j) {     // Ws: 32x128 floats = 1024 float4
            int idx = tid + j * 256;
            int k = idx >> 5, q = (idx & 31) * 4;
            async_ld16(lds_addr(&Ws[buf][k][q]),
                       W + (size_t)(kc + k) * HIDDEN + q);
        }
    };

    issue(0, 0);
    v8f c = {};

    for (int ch = 0; ch < 4; ++ch) {
        const int cur = ch & 1;
        wait_async0();                    // my async copies landed in LDS
        __syncthreads();                  // everyone's landed; prev compute done
        if (ch < 3) issue(1 - cur, (ch + 1) * 32);

#pragma unroll
        for (int k0 = 0; k0 < 32; k0 += 4) {
            const int kk = k0 + kb;
            v2f ah = *(const v2f*)&Hs[cur][m][kk];
            v2f aa = *(const v2f*)&Ag[cur][m][kk];
            v2f a  = ah + aa;             // fused (h + agg)
            v2f b;
            b.x = Ws[cur][kk][nbase + m];       // B[k][n]
            b.y = Ws[cur][kk + 1][nbase + m];   // B[k+1][n]
            c = __builtin_amdgcn_wmma_f32_16x16x4_f32(
                    /*neg_a=*/false, a, /*neg_b=*/false, b,
                    /*c_mod=*/(short)0, c, /*reuse_a=*/false, /*reuse_b=*/false);
        }
    }

    // epilogue: C/D layout -> VGPR r holds (row r + half*8, col = lane%16)
    int col = nbase + m;
    float bv = bias[col];
#pragma unroll
    for (int r = 0; r < 8; ++r) {
        int row = row0 + r + half * 8;
        hout[(size_t)row * HIDDEN + col] = fmaxf(c[r] + bv, 0.0f);
    }
}

// ---------------- global mean pool ----------------
__global__ void pool_accum(const float* __restrict__ h, const int* __restrict__ batch,
                           float* __restrict__ pooled, float* __restrict__ counts) {
    int n = blockIdx.x;
    int c = threadIdx.x;   // 128
    int g = batch[n];
    atomicAdd(&pooled[g * HIDDEN + c], h[(size_t)n * HIDDEN + c]);
    if (c == 0) atomicAdd(&counts[g], 1.0f);
}

__global__ void pool_mean(float* __restrict__ pooled, const float* __restrict__ counts) {
    int g = blockIdx.x;
    int c = threadIdx.x;
    pooled[g * HIDDEN + c] /= fmaxf(counts[g], 1.0f);
}

// ---------------- output MLP ----------------
__global__ void mlp1(const float* __restrict__ pooled, const float* __restrict__ W,
                     const float* __restrict__ b, float* __restrict__ hid) {
    int g = blockIdx.x;
    int j = threadIdx.x;   // 128
    float acc = b[j];
    for (int k = 0; k < HIDDEN; ++k)
        acc += pooled[g * HIDDEN + k] * W[k * HIDDEN + j];
    hid[g * HIDDEN + j] = fmaxf(acc, 0.0f);
}

__global__ void mlp2(const float* __restrict__ hid, const float* __restrict__ W,
                     const float* __restrict__ b, float* __restrict__ out) {
    int g = blockIdx.y;
    int o = blockIdx.x * blockDim.x + threadIdx.x;   // 512 cols
    float acc = b[o];
    for (int k = 0; k < HIDDEN; ++k)
        acc += hid[g * HIDDEN + k] * W[k * OUT_DIM + o];
    out[g * OUT_DIM + o] = acc;
}

extern "C" void kernel_launch(void* const* d_in, const int* in_sizes, int n_in,
                              void* d_out, int out_size, void* d_ws, size_t ws_size,
                              hipStream_t stream) {
    (void)in_sizes; (void)n_in; (void)out_size; (void)ws_size;

    const float* x     = (const float*)d_in[0];
    const int*   ei    = (const int*)d_in[1];   // [2, E] int32
    const int*   batch = (const int*)d_in[2];
    const float* pw    = (const float*)d_in[3];
    const float* pb    = (const float*)d_in[4];
    const float* cw    = (const float*)d_in[5]; // [3,128,128]
    const float* cb    = (const float*)d_in[6]; // [3,128]
    const float* w1    = (const float*)d_in[7];
    const float* b1    = (const float*)d_in[8];
    const float* w2    = (const float*)d_in[9];
    const float* b2    = (const float*)d_in[10];
    float* out = (float*)d_out;

    // workspace carve-up
    char* ws = (char*)d_ws;
    const size_t szH = (size_t)N_NODES * HIDDEN * sizeof(float);
    float* hA     = (float*)ws; ws += szH;
    float* hB     = (float*)ws; ws += szH;
    float* agg    = (float*)ws; ws += szH;
    float* pooled = (float*)ws; ws += (size_t)N_GRAPHS * HIDDEN * sizeof(float);
    float* counts = (float*)ws; ws += (size_t)N_GRAPHS * sizeof(float);
    float* hid    = (float*)ws;

    const int* src = ei;
    const int* dst = ei + N_EDGES;

    node_proj<<<N_NODES, HIDDEN, 0, stream>>>(x, pw, pb, hA);

    float* hin = hA;
    float* hout = hB;
    for (int l = 0; l < NUM_LAYERS; ++l) {
        const int nAgg = N_NODES * HIDDEN;
        zero_f32<<<(nAgg + 255) / 256, 256, 0, stream>>>(agg, nAgg);
        edge_scatter<<<(N_EDGES * 32 + 255) / 256, 256, 0, stream>>>(src, dst, hin, agg);
        gin_gemm<<<N_NODES / 16, 256, 0, stream>>>(hin, agg,
                                                   cw + (size_t)l * HIDDEN * HIDDEN,
                                                   cb + (size_t)l * HIDDEN, hout);
        float* t = hin; hin = hout; hout = t;
    }
    // final node features in hin

    zero_f32<<<(N_GRAPHS * HIDDEN + 255) / 256, 256, 0, stream>>>(pooled, N_GRAPHS * HIDDEN);
    zero_f32<<<1, 256, 0, stream>>>(counts, N_GRAPHS);
    pool_accum<<<N_NODES, HIDDEN, 0, stream>>>(hin, batch, pooled, counts);
    pool_mean<<<N_GRAPHS, HIDDEN, 0, stream>>>(pooled, counts);
    mlp1<<<N_GRAPHS, HIDDEN, 0, stream>>>(pooled, w1, b1, hid);
    mlp2<<<dim3(OUT_DIM / 256, N_GRAPHS), 256, 0, stream>>>(hid, w2, b2, out);
}